// ColorProcessingLossV2_188978561110
// MI455X (gfx1250) — compile-verified
//
// ColorProcessingLossV2 for MI455X (gfx1250).
//
// Memory-bound: ~180 MiB total traffic -> ~8 us at 23.3 TB/s. Two passes
// (global min/max is a serial dependency). Pass kernels stream f32 planes as
// b128 loads and stage the int32 label plane through LDS with the Tensor
// Data Mover (TENSORcnt double buffering). Final reduction of per-block
// partials uses v_wmma_f32_16x16x4_f32 (ones-matrix row-sum identity).

#include <hip/hip_runtime.h>
#include <cstdint>

#define TPB    256
#define CHUNK  1024          // pixels per TDM chunk (4 KiB of int32 labels)
#define NBLK   1024          // fixed grid for the streaming kernels
#define HWSH   19            // log2(512*1024) : pixels per image plane

typedef unsigned int u32x4 __attribute__((ext_vector_type(4)));
typedef int          i32x4 __attribute__((ext_vector_type(4)));
typedef int          i32x8 __attribute__((ext_vector_type(8)));
typedef float        v2f   __attribute__((ext_vector_type(2)));
typedef float        v8f   __attribute__((ext_vector_type(8)));

// per-class illumination constants, pre-divided by 255 (compile-time folded)
__device__ __constant__ float c_night[19] = {
  (float)(76.5113984140019/255.0),  (float)(76.23163212875781/255.0),
  (float)(60.90662084364415/255.0), (float)(69.06930071129905/255.0),
  (float)(69.63671393061327/255.0), (float)(73.11413822794262/255.0),
  (float)(140.7827781957324/255.0), (float)(116.29554873008291/255.0),
  (float)(46.23329954488532/255.0), (float)(57.839322341112386/255.0),
  (float)(32.61465346757989/255.0), (float)(57.4385179294615/255.0),
  (float)(62.234896087294814/255.0),(float)(90.90285758569436/255.0),
  (float)(91.99610158117673/255.0), (float)(91.82209397173472/255.0),
  (float)(94.06478985576457/255.0), (float)(74.6924145472464/255.0),
  (float)(69.15034088822232/255.0)
};
__device__ __constant__ float c_city[19] = {
  (float)(86.46051320057052/255.0), (float)(79.37014543897092/255.0),
  (float)(95.30679177391578/255.0), (float)(71.11888521745776/255.0),
  (float)(75.57026559270716/255.0), (float)(77.90493757655786/255.0),
  (float)(74.77466800282637/255.0), (float)(88.27701037425895/255.0),
  (float)(57.685269557270146/255.0),(float)(72.71472387765841/255.0),
  (float)(229.9589238353863/255.0), (float)(66.9194012998903/255.0),
  (float)(60.42471796718752/255.0), (float)(76.8407421534007/255.0),
  (float)(74.98657626719087/255.0), (float)(73.56771430328095/255.0),
  (float)(123.92515568872523/255.0),(float)(68.93476495876828/255.0),
  (float)(76.0970460111028/255.0)
};

// order-preserving float <-> uint for deterministic integer atomic min/max
__device__ __forceinline__ unsigned fenc(float f) {
  unsigned u = __float_as_uint(f);
  return (u & 0x80000000u) ? ~u : (u | 0x80000000u);
}
__device__ __forceinline__ float fdec(unsigned e) {
  unsigned u = (e & 0x80000000u) ? (e & 0x7FFFFFFFu) : ~e;
  return __uint_as_float(u);
}

// TDM: DMA `n` int32 elements from global to LDS byte offset `lds_off`.
// D# built per CDNA5 ISA 8.3/8.4: 1-descriptor, 2D tensor (tile = n x 1),
// data_size = 4B, no gather/iterate/pad, groups 2-3 unused (zero).
// This toolchain's builtin takes the 6-arg form:
//   (u32x4 g0, i32x8 g1, i32x4 g2, i32x4 g3, i32x8 extra, i32 cpol)
__device__ __forceinline__ void tdm_load_i32(const int* gsrc, unsigned lds_off,
                                             unsigned n) {
  const unsigned long long ga = (unsigned long long)(uintptr_t)gsrc;
  u32x4 g0;
  g0[0] = 1u;                                                  // count=1, user
  g0[1] = lds_off;                                             // lds_addr
  g0[2] = (unsigned)ga;                                        // gaddr[31:0]
  g0[3] = ((unsigned)(ga >> 32) & 0x01FFFFFFu) | 0x80000000u;  // gaddr[56:32]|type=2
  i32x8 g1;
  g1[0] = (int)(2u << 16);                           // data_size = 4 bytes
  g1[1] = (int)((n & 0xFFFFu) << 16);                // tensor_dim0[15:0]
  g1[2] = (int)(((n >> 16) & 0xFFFFu) | (1u << 16)); // td0[31:16] | td1[15:0]=1
  g1[3] = (int)((n & 0xFFFFu) << 16);                // td1[31:16]=0 | tile_dim0=n
  g1[4] = 1;                                         // tile_dim1=1, tile_dim2=0
  g1[5] = (int)n;                                    // tensor_dim0_stride
  g1[6] = 0;
  g1[7] = 0;
  const i32x4 z4 = {0, 0, 0, 0};
  const i32x8 z8 = {0, 0, 0, 0, 0, 0, 0, 0};
  __builtin_amdgcn_tensor_load_to_lds(g0, g1, z4, z4, z8, 0);
}

__global__ void k_init(unsigned* __restrict__ mmx) {
  if (threadIdx.x == 0) {
    mmx[0] = 0xFFFFFFFFu;  // min identity (encoded)
    mmx[1] = 0u;           // max identity (encoded)
  }
}

// Pass 1: global min/max of illum = max_c(org) - NIGHT[label]
__global__ __launch_bounds__(TPB) void k_minmax(const float* __restrict__ org,
                                                const int* __restrict__ lab,
                                                unsigned* __restrict__ mmx,
                                                int np) {
  __shared__ __align__(16) int lbuf[2][CHUNK];
  __shared__ float s_night[19];
  __shared__ float s_red[2][8];
  if (threadIdx.x < 19) s_night[threadIdx.x] = c_night[threadIdx.x];

  const int wave = (int)threadIdx.x >> 5;
  const int npb  = np >> 10;    // pixels per block (grid = NBLK = 1024)
  const int nch  = npb >> 10;   // chunks per block
  const int base = (int)blockIdx.x * npb;
  const unsigned lds0 = (unsigned)(uintptr_t)&lbuf[0][0];

  if (wave == 0) tdm_load_i32(lab + base, lds0, CHUNK);

  float mn = 3.0e38f, mx = -3.0e38f;
  for (int ch = 0; ch < nch; ++ch) {
    if (wave == 0) __builtin_amdgcn_s_wait_tensorcnt((short)0);
    __syncthreads();  // chunk `ch` visible to all waves; prior reads done
    if (wave == 0 && ch + 1 < nch)
      tdm_load_i32(lab + base + (ch + 1) * CHUNK,
                   lds0 + (unsigned)(((ch + 1) & 1) * (CHUNK * 4)), CHUNK);

    const int p0 = base + ch * CHUNK + (int)threadIdx.x * 4;
    const int b  = p0 >> HWSH;
    const int hw = p0 & ((1 << HWSH) - 1);
    const size_t ob = ((size_t)(b * 3) << HWSH) + (size_t)hw;
    const float4 o0 = *(const float4*)(org + ob);
    const float4 o1 = *(const float4*)(org + ob + ((size_t)1 << HWSH));
    const float4 o2 = *(const float4*)(org + ob + ((size_t)2 << HWSH));
    const int4 lb = *(const int4*)&lbuf[ch & 1][threadIdx.x * 4];
#define PIXMM(F)                                                      \
    { float il = fmaxf(fmaxf(o0.F, o1.F), o2.F) - s_night[lb.F];      \
      mn = fminf(mn, il); mx = fmaxf(mx, il); }
    PIXMM(x) PIXMM(y) PIXMM(z) PIXMM(w)
#undef PIXMM
  }

  for (int off = 16; off > 0; off >>= 1) {
    mn = fminf(mn, __shfl_xor(mn, off, 32));
    mx = fmaxf(mx, __shfl_xor(mx, off, 32));
  }
  if ((threadIdx.x & 31) == 0) { s_red[0][wave] = mn; s_red[1][wave] = mx; }
  __syncthreads();
  if (threadIdx.x == 0) {
    float bmn = s_red[0][0], bmx = s_red[1][0];
    for (int w = 1; w < 8; ++w) {
      bmn = fminf(bmn, s_red[0][w]);
      bmx = fmaxf(bmx, s_red[1][w]);
    }
    atomicMin(&mmx[0], fenc(bmn));
    atomicMax(&mmx[1], fenc(bmx));
  }
}

// Pass 2: normalize, gamma (gmax==1 after min-max norm), add CITY, clip,
// multiply reflectance, per-channel partial sums per block.
__global__ __launch_bounds__(TPB) void k_bright(const float* __restrict__ refl,
                                                const float* __restrict__ org,
                                                const int* __restrict__ lab,
                                                const unsigned* __restrict__ mmx,
                                                float* __restrict__ part,
                                                int np) {
  __shared__ __align__(16) int lbuf[2][CHUNK];
  __shared__ float s_night[19];
  __shared__ float s_city[19];
  __shared__ float s_sum[3][8];
  if (threadIdx.x < 19) {
    s_night[threadIdx.x] = c_night[threadIdx.x];
    s_city[threadIdx.x]  = c_city[threadIdx.x];
  }

  const float mn = fdec(mmx[0]);
  const float mx = fdec(mmx[1]);
  const float d  = mx - mn;
  const float inv = (d > 0.0f) ? (1.0f / d) : 0.0f;

  const int wave = (int)threadIdx.x >> 5;
  const int npb  = np >> 10;
  const int nch  = npb >> 10;
  const int base = (int)blockIdx.x * npb;
  const unsigned lds0 = (unsigned)(uintptr_t)&lbuf[0][0];

  if (wave == 0) tdm_load_i32(lab + base, lds0, CHUNK);

  float s0 = 0.0f, s1 = 0.0f, s2 = 0.0f;
  for (int ch = 0; ch < nch; ++ch) {
    if (wave == 0) __builtin_amdgcn_s_wait_tensorcnt((short)0);
    __syncthreads();
    if (wave == 0 && ch + 1 < nch)
      tdm_load_i32(lab + base + (ch + 1) * CHUNK,
                   lds0 + (unsigned)(((ch + 1) & 1) * (CHUNK * 4)), CHUNK);

    const int p0 = base + ch * CHUNK + (int)threadIdx.x * 4;
    const int b  = p0 >> HWSH;
    const int hw = p0 & ((1 << HWSH) - 1);
    const size_t ob = ((size_t)(b * 3) << HWSH) + (size_t)hw;
    const float4 o0 = *(const float4*)(org + ob);
    const float4 o1 = *(const float4*)(org + ob + ((size_t)1 << HWSH));
    const float4 o2 = *(const float4*)(org + ob + ((size_t)2 << HWSH));
    const float4 r0 = *(const float4*)(refl + ob);
    const float4 r1 = *(const float4*)(refl + ob + ((size_t)1 << HWSH));
    const float4 r2 = *(const float4*)(refl + ob + ((size_t)2 << HWSH));
    const int4 lb = *(const int4*)&lbuf[ch & 1][threadIdx.x * 4];
#define PIX(F)                                                          \
    { float il = fmaxf(fmaxf(o0.F, o1.F), o2.F) - s_night[lb.F];        \
      float t  = fminf(fmaxf((il - mn) * inv, 0.0f), 1.0f);             \
      t = __powf(t, 0.4f);                                              \
      float wgt = fminf(fmaxf(t + s_city[lb.F], 0.0f), 1.0f);           \
      s0 += r0.F * wgt; s1 += r1.F * wgt; s2 += r2.F * wgt; }
    PIX(x) PIX(y) PIX(z) PIX(w)
#undef PIX
  }

  for (int off = 16; off > 0; off >>= 1) {
    s0 += __shfl_xor(s0, off, 32);
    s1 += __shfl_xor(s1, off, 32);
    s2 += __shfl_xor(s2, off, 32);
  }
  if ((threadIdx.x & 31) == 0) {
    s_sum[0][wave] = s0; s_sum[1][wave] = s1; s_sum[2][wave] = s2;
  }
  __syncthreads();
  if (threadIdx.x == 0) {
    float t0 = 0.0f, t1 = 0.0f, t2 = 0.0f;
    for (int w = 0; w < 8; ++w) { t0 += s_sum[0][w]; t1 += s_sum[1][w]; t2 += s_sum[2][w]; }
    part[0 * NBLK + blockIdx.x] = t0;
    part[1 * NBLK + blockIdx.x] = t1;
    part[2 * NBLK + blockIdx.x] = t2;
  }
}

// Final: reduce 1024 partials/channel on the matrix pipe.
// With B == ones(4x16): D[m][n] += sum_k A[m][k]. Each wmma folds 64 floats
// into C; 16 wmmas fold all 1024; total = sum_m D[m][0] = lane0 + lane16 sums.
// One wave of 32 threads, uniform flow => EXEC all ones as WMMA requires.
__global__ void k_final(const float* __restrict__ part, float* __restrict__ out,
                        int np) {
  const int lane = (int)threadIdx.x;  // blockDim.x == 32
  v2f ones; ones[0] = 1.0f; ones[1] = 1.0f;
  float S[3];
  for (int c = 0; c < 3; ++c) {
    v8f acc = {};
    const float* P = part + c * NBLK;
    for (int i = 0; i < NBLK / 64; ++i) {
      v2f a;
      a[0] = P[i * 64 + lane];
      a[1] = P[i * 64 + 32 + lane];
      acc = __builtin_amdgcn_wmma_f32_16x16x4_f32(
          /*neg_a=*/false, a, /*neg_b=*/false, ones,
          /*c_mod=*/(short)0, acc, /*reuse_a=*/false, /*reuse_b=*/false);
    }
    float s = acc[0] + acc[1] + acc[2] + acc[3] +
              acc[4] + acc[5] + acc[6] + acc[7];
    S[c] = __shfl(s, 0, 32) + __shfl(s, 16, 32);
  }
  if (lane == 0) {
    const float invn = 1.0f / (float)np;
    const float r = S[0] * invn, g = S[1] * invn, b = S[2] * invn;
    out[0] = (r - g) * (r - g) + (r - b) * (r - b) + (g - b) * (g - b);
  }
}

extern "C" void kernel_launch(void* const* d_in, const int* in_sizes, int n_in,
                              void* d_out, int out_size, void* d_ws, size_t ws_size,
                              hipStream_t stream) {
  (void)n_in; (void)out_size; (void)ws_size;
  const float* refl = (const float*)d_in[0];
  const float* org  = (const float*)d_in[1];
  const int*   lab  = (const int*)d_in[2];
  float* out = (float*)d_out;
  const int np = in_sizes[2];  // B*H*W = 4,194,304 pixels

  unsigned* mmx = (unsigned*)d_ws;           // [0]=min(enc), [1]=max(enc)
  float* part = (float*)d_ws + 64;           // 3*NBLK partial sums (256B off)

  k_init  <<<1,    1,   0, stream>>>(mmx);
  k_minmax<<<NBLK, TPB, 0, stream>>>(org, lab, mmx, np);
  k_bright<<<NBLK, TPB, 0, stream>>>(refl, org, lab, mmx, part, np);
  k_final <<<1,    32,  0, stream>>>(part, out, np);
}